// EquiLinearRegToReg_44023414784206
// MI455X (gfx1250) — compile-verified
//
#include <hip/hip_runtime.h>

// CDNA5 / gfx1250 — fp32 WMMA (V_WMMA_F32_16X16X4_F32), wave32.
typedef float v2f __attribute__((ext_vector_type(2)));
typedef float v4f __attribute__((ext_vector_type(4)));
typedef float v8f __attribute__((ext_vector_type(8)));

#define IN_FEAT  256
#define OUT_FEAT 256
#define KCONV    16
#define BPROWS   4096   // BATCH(8) * NUM_PART(512)

// One wave computes a 16(bp) x 16(j) x 16(y) output macro-tile.
// out[bp, j, y] = sum_{i,x} W[i, j, (y-x)&15] * F[bp, i, x]
__global__ __launch_bounds__(256)
void EquiLinearRegToReg_wmma(const float* __restrict__ F,   // [4096][256][16]
                             const float* __restrict__ W,   // [256][256][16]
                             float* __restrict__ O)         // [4096][256][16]
{
    const int wave = blockIdx.x * (blockDim.x >> 5) + (threadIdx.x >> 5);
    const int lane = threadIdx.x & 31;
    const int jT   = wave & 15;    // 16 j-tiles
    const int bpT  = wave >> 4;    // 256 bp-tiles
    const int row  = lane & 15;    // M / N index within tile
    const int hi   = lane >> 4;    // selects K pair {0,1} vs {2,3}

    const int bp0 = bpT * 16;
    const int j0  = jT * 16;

    v8f acc[16] = {};  // acc[y] : 16x16 f32 tile, 8 VGPRs each -> 128 VGPRs

    // A fragment source: F[(bp0+row)][i0 + 2*hi + {0,1}][x], x = 0..15 contiguous
    const float* aBase = F + ((size_t)(bp0 + row) * IN_FEAT + 2 * hi) * KCONV;
    // B fragment source: W[i0 + 2*hi + {0,1}][j0+row][s], s = 0..15 contiguous
    const float* bBase = W + ((size_t)(2 * hi) * OUT_FEAT + (j0 + row)) * KCONV;

    for (int i0 = 0; i0 < IN_FEAT; i0 += 4) {
        const float* ap = aBase + (size_t)i0 * KCONV;
        const float* bp = bBase + (size_t)i0 * OUT_FEAT * KCONV;

        v4f a0[4], a1[4], b0[4], b1[4];
#pragma unroll
        for (int q = 0; q < 4; ++q) {
            a0[q] = *(const v4f*)(ap + 4 * q);                           // i = i0+2*hi,   x=4q..4q+3
            a1[q] = *(const v4f*)(ap + KCONV + 4 * q);                   // i = i0+2*hi+1
            b0[q] = *(const v4f*)(bp + 4 * q);                           // i = i0+2*hi,   s=4q..4q+3
            b1[q] = *(const v4f*)(bp + (size_t)OUT_FEAT * KCONV + 4 * q);// i = i0+2*hi+1
        }

#pragma unroll
        for (int x = 0; x < 16; ++x) {
            v2f A;
            A[0] = a0[x >> 2][x & 3];   // K row 2*hi
            A[1] = a1[x >> 2][x & 3];   // K row 2*hi+1
#pragma unroll
            for (int y = 0; y < 16; ++y) {
                const int s = (y - x) & 15;
                v2f B;
                B[0] = b0[s >> 2][s & 3];
                B[1] = b1[s >> 2][s & 3];
                // D = A(16x4) * B(4x16) + C  (f32)
                acc[y] = __builtin_amdgcn_wmma_f32_16x16x4_f32(
                    /*neg_a=*/false, A, /*neg_b=*/false, B,
                    /*c_mod=*/(short)0, acc[y],
                    /*reuse_a=*/false, /*reuse_b=*/false);
            }
        }
    }

    // C/D layout: lane holds N = row, rows M = v + 8*hi (v = VGPR 0..7).
    // For fixed (bp, j), the 16 y-values are contiguous in memory.
#pragma unroll
    for (int v = 0; v < 8; ++v) {
        float* op = O + ((size_t)(bp0 + v + 8 * hi) * OUT_FEAT + (j0 + row)) * KCONV;
#pragma unroll
        for (int q = 0; q < 4; ++q) {
            v4f t;
            t[0] = acc[4 * q + 0][v];
            t[1] = acc[4 * q + 1][v];
            t[2] = acc[4 * q + 2][v];
            t[3] = acc[4 * q + 3][v];
            *(v4f*)(op + 4 * q) = t;
        }
    }
}

extern "C" void kernel_launch(void* const* d_in, const int* in_sizes, int n_in,
                              void* d_out, int out_size, void* d_ws, size_t ws_size,
                              hipStream_t stream) {
    const float* field   = (const float*)d_in[0];  // (8, 512, 256, 16) f32
    const float* weights = (const float*)d_in[1];  // (256, 256, 16)    f32
    float* out = (float*)d_out;                    // (8, 512, 256, 16) f32

    // 4096 waves total: 256 bp-tiles x 16 j-tiles. 8 waves / 256-thread block.
    dim3 grid(512), block(256);
    EquiLinearRegToReg_wmma<<<grid, block, 0, stream>>>(field, weights, out);
}